// HydrologyModel_18537078849622
// MI455X (gfx1250) — compile-verified
//
#include <hip/hip_runtime.h>
#include <stdint.h>
#include <stddef.h>

// ---------------------------------------------------------------------------
// EXP-HYDRO on MI455X (gfx1250).
// Bandwidth-bound streaming recurrence (~352 MB -> ~15us floor @ 23.3 TB/s).
//  - Phase 1: serial state scan (256 waves only) -> deep 11-slot async
//    global->LDS ring (60 ASYNCcnt ops/wave = 30KB/wave in flight) to beat
//    latency-boundedness; writes state checkpoints every CHUNK steps.
//  - Phase 2: (T/CHUNK x B/256) blocks recompute from checkpoints and emit
//    all 8 fluxes with non-temporal stores (output never re-read; forcing
//    re-reads hit the 192MB L2 which holds the whole 96MB forcing set).
// ---------------------------------------------------------------------------

#define NEARZERO 1e-6f
#define BT      256                    // basins per block (8 wave32)
#define TILE_T  8                      // timesteps per LDS tile
#define CHUNK   64                     // checkpoint spacing
#define TILE_FLOATS (3 * TILE_T * BT)  // floats per tile (24 KB)

#ifndef __has_builtin
#define __has_builtin(x) 0
#endif

#if defined(__HIP_DEVICE_COMPILE__) && __has_builtin(__builtin_amdgcn_global_load_async_to_lds_b128)
#define HAVE_ASYNC_LDS 1
#else
#define HAVE_ASYNC_LDS 0
#endif

#if HAVE_ASYNC_LDS
typedef int v4i __attribute__((ext_vector_type(4)));
typedef __attribute__((address_space(1))) v4i gv4i;   // global int4
typedef __attribute__((address_space(3))) v4i lv4i;   // LDS int4
#endif

// One lane moves 16 B (4 basins) global -> LDS, tracked by ASYNCcnt.
__device__ __forceinline__ void async_b128(const float* g, float* l) {
#if HAVE_ASYNC_LDS
    __builtin_amdgcn_global_load_async_to_lds_b128((gv4i*)g, (lv4i*)l, 0, 0);
#else
    *(float4*)l = *(const float4*)g;
#endif
}

template <int N>
__device__ __forceinline__ void wait_async() {
#if HAVE_ASYNC_LDS
#if __has_builtin(__builtin_amdgcn_s_wait_asynccnt)
    __builtin_amdgcn_s_wait_asynccnt(N);
#else
    asm volatile("s_wait_asynccnt %0" ::"n"(N) : "memory");
#endif
#endif
}

struct HParams { float f, Smax, Qmax, Df, Tmax, Tmin, invSmax; };

__device__ __forceinline__ HParams load_params(const float* __restrict__ sp, int b) {
    const float lo[6] = {0.0f, 100.0f, 10.0f, 0.0f, 0.0f, -3.0f};
    const float hi[6] = {0.1f, 1500.0f, 50.0f, 5.0f, 3.0f,  0.0f};
    float v[6];
#pragma unroll
    for (int i = 0; i < 6; ++i) {
        float x = sp[(size_t)b * 6 + i];
        float s = 1.0f / (1.0f + __expf(-x));     // sigmoid
        v[i] = lo[i] + (hi[i] - lo[i]) * s;
    }
    HParams p;
    p.f = v[0]; p.Smax = v[1]; p.Qmax = v[2];
    p.Df = v[3]; p.Tmax = v[4]; p.Tmin = v[5];
    p.invSmax = 1.0f / v[1];
    return p;
}

// Stage one tile [3][TILE_T][BT] into LDS: 256 threads = 4 rows x 64 b128 lanes,
// 6 async instructions per thread (per wave: 6 ASYNCcnt increments, 3 KB).
__device__ __forceinline__ void issue_tile(const float* __restrict__ prcp,
                                           const float* __restrict__ temp,
                                           const float* __restrict__ pet,
                                           int t0, int b0, int B,
                                           float* __restrict__ slot, int tid) {
    const int row = tid >> 6;          // 0..3
    const int sub = tid & 63;          // 0..63 -> floats [sub*4 .. sub*4+3]
#pragma unroll
    for (int i = 0; i < (3 * TILE_T) / 4; ++i) {
        const int p   = row + 4 * i;               // 0..23
        const int arr = p >> 3;                    // p / TILE_T (TILE_T==8)
        const int tt  = p & (TILE_T - 1);          // p % TILE_T
        const float* src = (arr == 0) ? prcp : ((arr == 1) ? temp : pet);
        const float* g = src + (size_t)(t0 + tt) * (size_t)B + (size_t)(b0 + (sub << 2));
        float* l = slot + (arr * TILE_T + tt) * BT + (sub << 2);
        async_b128(g, l);
    }
}

template <bool WRITE_CKPT, bool WRITE_OUT>
__device__ __forceinline__ void compute_tile(const float* __restrict__ cb, int tid,
                                             int t0, int b, size_t Bs, const HParams& P,
                                             float& snow, float& soil,
                                             float* __restrict__ out,
                                             float* __restrict__ ckpt) {
    if (WRITE_CKPT && (t0 % CHUNK == 0)) {
        const size_t c = (size_t)(t0 / CHUNK);
        ckpt[(c * 2 + 0) * Bs + b] = snow;
        ckpt[(c * 2 + 1) * Bs + b] = soil;
    }
#pragma unroll
    for (int tt = 0; tt < TILE_T; ++tt) {
        const float prcp_t = cb[(0 * TILE_T + tt) * BT + tid];
        const float temp_t = cb[(1 * TILE_T + tt) * BT + tid];
        const float pet_t  = cb[(2 * TILE_T + tt) * BT + tid];

        const float snowfall    = (temp_t < P.Tmin) ? prcp_t : 0.0f;
        const float rainfall    = prcp_t - snowfall;
        const float melt        = fminf(snow, P.Df * fmaxf(temp_t - P.Tmax, 0.0f));
        const float evap        = pet_t * fminf(soil * P.invSmax, 1.0f);
        const float baseflow    = P.Qmax * __expf(-P.f * fmaxf(P.Smax - soil, 0.0f));
        const float surfaceflow = fmaxf(soil - P.Smax, 0.0f);
        const float d_snow      = snowfall - melt;
        const float d_soil      = rainfall + melt - evap - baseflow - surfaceflow;

        snow = fmaxf(snow + d_snow, NEARZERO);
        soil = fmaxf(soil + d_soil, NEARZERO);

        if (WRITE_OUT) {
            // out[t][f][b]: contiguous in b per wave -> perfectly coalesced.
            // Output is never re-read: non-temporal to spare L2 for forcing.
            float* o = out + ((size_t)(t0 + tt) * 8) * Bs + (size_t)b;
            __builtin_nontemporal_store(snowfall,    o + 0 * Bs);
            __builtin_nontemporal_store(rainfall,    o + 1 * Bs);
            __builtin_nontemporal_store(melt,        o + 2 * Bs);
            __builtin_nontemporal_store(evap,        o + 3 * Bs);
            __builtin_nontemporal_store(baseflow,    o + 4 * Bs);
            __builtin_nontemporal_store(surfaceflow, o + 5 * Bs);
            __builtin_nontemporal_store(d_snow,      o + 6 * Bs);
            __builtin_nontemporal_store(d_soil,      o + 7 * Bs);
        }
    }
}

// Ring-buffered async pipeline over NBUF LDS tile slots.
// Steady state keeps NBUF-1 tiles in flight; async loads complete in order
// per wave, so tile k is resident once ASYNCcnt <= 6*(NBUF-2).
template <bool LOAD_STATE, bool WRITE_CKPT, bool WRITE_OUT, int NBUF>
__global__ __launch_bounds__(BT) void hydro_kernel(const float* __restrict__ prcp,
                                                   const float* __restrict__ temp,
                                                   const float* __restrict__ pet,
                                                   const float* __restrict__ sp,
                                                   float* __restrict__ out,
                                                   float* __restrict__ ckpt,
                                                   int B, int tSpan) {
    static_assert(6 * (NBUF - 1) <= 63, "ASYNCcnt is 6-bit");
    extern __shared__ float dynbuf[];   // NBUF * TILE_FLOATS floats

    const int tid    = (int)threadIdx.x;
    const int b0     = (int)blockIdx.x * BT;
    const int b      = b0 + tid;
    const int tBegin = (int)blockIdx.y * tSpan;
    const size_t Bs  = (size_t)B;

    const HParams P = load_params(sp, b);

    float snow, soil;
    if (LOAD_STATE) {
        const size_t c0 = (size_t)(tBegin / CHUNK);
        snow = ckpt[(c0 * 2 + 0) * Bs + b];
        soil = ckpt[(c0 * 2 + 1) * Bs + b];
    } else {
        snow = 0.0f;
        soil = 0.0f;
    }

    const int ntiles = tSpan / TILE_T;

    // Prologue: fill pipeline with up to NBUF-1 tiles.
    const int pre = (ntiles < NBUF - 1) ? ntiles : (NBUF - 1);
    for (int k = 0; k < pre; ++k)
        issue_tile(prcp, temp, pet, tBegin + k * TILE_T, b0, B,
                   dynbuf + (size_t)(k % NBUF) * TILE_FLOATS, tid);

    // Steady state.
    const int steady = ntiles - pre;
    for (int k = 0; k < steady; ++k) {
        wait_async<6 * (NBUF - 2)>();   // my share of tile k has landed
        __syncthreads();                // everyone's share landed; slot (k-1)%NBUF free
        const int kp = k + NBUF - 1;
        issue_tile(prcp, temp, pet, tBegin + kp * TILE_T, b0, B,
                   dynbuf + (size_t)(kp % NBUF) * TILE_FLOATS, tid);
        compute_tile<WRITE_CKPT, WRITE_OUT>(dynbuf + (size_t)(k % NBUF) * TILE_FLOATS,
                                            tid, tBegin + k * TILE_T, b, Bs, P,
                                            snow, soil, out, ckpt);
    }

    // Epilogue: everything issued; drain with no further barriers.
    wait_async<0>();
    __syncthreads();
    for (int k = steady; k < ntiles; ++k)
        compute_tile<WRITE_CKPT, WRITE_OUT>(dynbuf + (size_t)(k % NBUF) * TILE_FLOATS,
                                            tid, tBegin + k * TILE_T, b, Bs, P,
                                            snow, soil, out, ckpt);
}

extern "C" void kernel_launch(void* const* d_in, const int* in_sizes, int n_in,
                              void* d_out, int out_size, void* d_ws, size_t ws_size,
                              hipStream_t stream) {
    const float* prcp = (const float*)d_in[0];
    const float* temp = (const float*)d_in[1];
    const float* pet  = (const float*)d_in[2];
    const float* sp   = (const float*)d_in[3];
    float* out = (float*)d_out;

    const int B = in_sizes[3] / 6;          // static_params_norm: [B, 6]
    const int T = in_sizes[0] / B;          // prcp: [T, B]
    const int nbasinBlocks = B / BT;
    const int nchunk = T / CHUNK;
    const size_t ckptBytes = (size_t)nchunk * 2 * (size_t)B * sizeof(float);

    constexpr int NB1 = 11;  // phase 1: 264 KB LDS ring, 60 async ops/wave in flight
    constexpr int NB2 = 4;   // phase 2 / fallback: 96 KB LDS ring
    const size_t shmem1 = (size_t)NB1 * TILE_FLOATS * sizeof(float);
    const size_t shmem2 = (size_t)NB2 * TILE_FLOATS * sizeof(float);

    const bool twoPhase = (d_ws != nullptr) && (ws_size >= ckptBytes) &&
                          (T % CHUNK == 0) && (B % BT == 0) && (CHUNK % TILE_T == 0);

    if (twoPhase) {
        float* ckpt = (float*)d_ws;
        // Phase 1: states only, checkpoints every CHUNK steps, deep async ring.
        hipLaunchKernelGGL(HIP_KERNEL_NAME(hydro_kernel<false, true, false, NB1>),
                           dim3(nbasinBlocks, 1), dim3(BT), shmem1, stream,
                           prcp, temp, pet, sp, (float*)nullptr, ckpt, B, T);
        // Phase 2: parallel recompute + flux emission.
        hipLaunchKernelGGL(HIP_KERNEL_NAME(hydro_kernel<true, false, true, NB2>),
                           dim3(nbasinBlocks, nchunk), dim3(BT), shmem2, stream,
                           prcp, temp, pet, sp, out, ckpt, B, CHUNK);
    } else {
        // Fallback: monolithic single pass.
        hipLaunchKernelGGL(HIP_KERNEL_NAME(hydro_kernel<false, false, true, NB2>),
                           dim3(nbasinBlocks, 1), dim3(BT), shmem2, stream,
                           prcp, temp, pet, sp, out, (float*)nullptr, B, T);
    }
}